// GRNN_3547642986522
// MI455X (gfx1250) — compile-verified
//
#include <hip/hip_runtime.h>
#include <stdint.h>

// ---------------------------------------------------------------------------
// GRNN / Kalman-filter scan for MI455X (gfx1250, wave32).
// Memory-bound streaming scan: stage dy / dy_hat tiles through LDS with the
// CDNA5 async global<->LDS copy engine (ASYNCcnt), double-buffered so the
// copies overlap the sequential per-row recurrence.
// ---------------------------------------------------------------------------

#define GLB_AS __attribute__((address_space(1)))
#define LDS_AS __attribute__((address_space(3)))

typedef int v4i __attribute__((ext_vector_type(4)));

__device__ __forceinline__ void async_ld_b128(const void* g, void* l) {
#if __has_builtin(__builtin_amdgcn_global_load_async_to_lds_b128)
  __builtin_amdgcn_global_load_async_to_lds_b128(
      (GLB_AS v4i*)const_cast<void*>(g), (LDS_AS v4i*)l, 0, 0);
#else
  unsigned loff = (unsigned)(uintptr_t)(LDS_AS void*)l;
  asm volatile("global_load_async_to_lds_b128 %0, %1, off"
               :: "v"(loff), "v"(g) : "memory");
#endif
}

__device__ __forceinline__ void async_st_b128(void* g, const void* l) {
#if __has_builtin(__builtin_amdgcn_global_store_async_from_lds_b128)
  __builtin_amdgcn_global_store_async_from_lds_b128(
      (GLB_AS v4i*)g, (LDS_AS v4i*)const_cast<void*>(l), 0, 0);
#else
  unsigned loff = (unsigned)(uintptr_t)(LDS_AS void*)const_cast<void*>(l);
  asm volatile("global_store_async_from_lds_b128 %0, %1, off"
               :: "v"(g), "v"(loff) : "memory");
#endif
}

__device__ __forceinline__ void wait_async0() {
#if __has_builtin(__builtin_amdgcn_s_wait_asynccnt)
  __builtin_amdgcn_s_wait_asynccnt(0);
#else
  asm volatile("s_wait_asynccnt 0" ::: "memory");
#endif
}

// Problem geometry
static constexpr int B_TOT = 4096;
static constexpr int T_TOT = 2048;
static constexpr int TC    = 32;              // timesteps per LDS tile
static constexpr int NT    = T_TOT / TC;      // 64 tiles
static constexpr int ROWS  = 64;              // batch rows per workgroup
static constexpr int NTHR  = 64;              // 2 waves (wave32)
static constexpr int ROWF  = TC * 2 + 4;      // 68 floats = 272B: 16B-aligned row stride

extern "C" __global__ __launch_bounds__(NTHR)
void grnn_filter_kernel(const float* __restrict__ dy,
                        const float* __restrict__ state,
                        const float* __restrict__ fin,
                        const float* __restrict__ Kp,
                        float* __restrict__ out)
{
  __shared__ float smem[(2 * ROWS + ROWS) * ROWF];   // 52,224 B
  float* inb0 = smem;
  float* inb1 = smem + ROWS * ROWF;
  float* outb = smem + 2 * ROWS * ROWF;

  const int tid   = (int)threadIdx.x;
  const int rbase = (int)blockIdx.x * ROWS;
  const int b     = rbase + tid;

  // Physics constants (folded from the reference)
  const float DT   = 0.001f;
  const float g    = 0.3f;
  const float hg   = 0.15f;                      // gamma/2
  const float w    = 6.283185f;
  const float cS   = 1.6970562748477141f;        // sqrt(4*eta*kappa)
  const float c2   = cS * cS;
  const float dcst = 5.25f;                      // gamma*(nth+0.5)+kappa
  const float cDT  = cS * DT;

  // kernel_params is a runtime input
  const float k00 = Kp[0], k01 = Kp[1], k10 = Kp[2], k11 = Kp[3];

  // carried state for this row
  float x0 = state[b * 6 + 0], x1 = state[b * 6 + 1];
  float vx = state[b * 6 + 2], vp = state[b * 6 + 3], cxp = state[b * 6 + 4];
  float tt = state[b * 6 + 5];
  float f0 = fin[b * 2 + 0], f1 = fin[b * 2 + 1];

  float* dyhat = out + B_TOT * 6;   // (B, T, 2) region of the output

  // Async-copy chunk mapping: each row of a tile is TC*2 floats = 256 B =
  // 16 chunks of 16 B. 64 threads * 16 iterations cover ROWS*16 chunks.
  const int sub = tid & 15;   // 16B chunk index within a row
  const int rw  = tid >> 4;   // row sub-index 0..3

  // Preload tile 0 into buffer 0
  for (int j = 0; j < ROWS / 4; ++j) {
    const int r = j * 4 + rw;
    const float* gsrc = dy + ((size_t)(rbase + r) * T_TOT) * 2 + sub * 4;
    async_ld_b128(gsrc, inb0 + r * ROWF + sub * 4);
  }

  for (int k = 0; k < NT; ++k) {
    wait_async0();        // tile k loaded (and tile k-1 stores drained)
    __syncthreads();      // make every wave's async LDS writes visible

    float* inb = (k & 1) ? inb1 : inb0;

    // Prefetch tile k+1 while we compute on tile k
    if (k + 1 < NT) {
      float* nxt = ((k + 1) & 1) ? inb1 : inb0;
      for (int j = 0; j < ROWS / 4; ++j) {
        const int r = j * 4 + rw;
        const float* gsrc =
            dy + (((size_t)(rbase + r) * T_TOT) + (size_t)(k + 1) * TC) * 2 + sub * 4;
        async_ld_b128(gsrc, nxt + r * ROWF + sub * 4);
      }
    }

    const float* inrow = inb + tid * ROWF;
    float* orow        = outb + tid * ROWF;

#pragma unroll
    for (int t = 0; t < TC; ++t) {
      const float dy0 = inrow[2 * t];       // dy[...,1] is provably unused
      const float yh0 = cDT * x0;           // dy_hat uses pre-update x

      // x update: M = A - c^2*[[vx,0],[cxp,0]]; xicov@dy = c*(vx,cxp)*dy0
      const float m00 = -hg - c2 * vx;
      const float m10 = -w  - c2 * cxp;
      float dx0 = (m00 * x0 + w * x1) * DT + (cS * vx) * dy0;
      float dx1 = (m10 * x0 - hg * x1) * DT + (cS * cxp) * dy0;

      // f update (K is runtime data), then PROJ_F @ fnew = (0, fnew0)
      const float f0n = f0 + (k00 * f0 + k01 * f1) * DT;
      const float f1n = f1 + (k10 * f0 + k11 * f1) * DT;
      dx1 += f0n * DT;

      // Riccati covariance update (data-independent)
      const float nvx  = vx  + DT * (-g * vx + 2.0f * w * cxp + dcst - c2 * vx * vx);
      const float nvp  = vp  + DT * (-g * vp - 2.0f * w * cxp + dcst - c2 * cxp * cxp);
      const float ncxp = cxp + DT * (w * (vp - vx) - g * cxp - c2 * vx * cxp);

      x0 += dx0; x1 += dx1;
      vx = nvx; vp = nvp; cxp = ncxp;
      f0 = f0n; f1 = f1n;
      tt += DT;

      orow[2 * t]     = yh0;
      orow[2 * t + 1] = 0.0f;
    }

    __syncthreads();      // out tile fully written before cross-row stores

    // Async-store the dy_hat tile back (coalesced per-row 256B chunks)
    for (int j = 0; j < ROWS / 4; ++j) {
      const int r = j * 4 + rw;
      float* gdst =
          dyhat + (((size_t)(rbase + r) * T_TOT) + (size_t)k * TC) * 2 + sub * 4;
      async_st_b128(gdst, outb + r * ROWF + sub * 4);
    }
  }

  // Final carried outputs: nstate (B,6) at offset 0, fnew (B,2) after dy_hat
  out[b * 6 + 0] = x0;
  out[b * 6 + 1] = x1;
  out[b * 6 + 2] = vx;
  out[b * 6 + 3] = vp;
  out[b * 6 + 4] = cxp;
  out[b * 6 + 5] = tt;
  float* fout = out + (size_t)B_TOT * 6 + (size_t)B_TOT * T_TOT * 2;
  fout[b * 2 + 0] = f0;
  fout[b * 2 + 1] = f1;

  wait_async0();  // drain final stores (s_endpgm also waits-idle)
}

extern "C" void kernel_launch(void* const* d_in, const int* in_sizes, int n_in,
                              void* d_out, int out_size, void* d_ws, size_t ws_size,
                              hipStream_t stream) {
  const float* dy = (const float*)d_in[0];
  const float* st = (const float*)d_in[1];
  const float* f  = (const float*)d_in[2];
  const float* K  = (const float*)d_in[3];
  float* out = (float*)d_out;
  (void)in_sizes; (void)n_in; (void)out_size; (void)d_ws; (void)ws_size;

  dim3 grid(B_TOT / ROWS);   // 64 workgroups -> spread across WGPs
  dim3 block(NTHR);          // 64 threads = 2 wave32 waves
  hipLaunchKernelGGL(grnn_filter_kernel, grid, block, 0, stream, dy, st, f, K, out);
}